// PostProcessor_83992380440892
// MI455X (gfx1250) — compile-verified
//
#include <hip/hip_runtime.h>
#include <stdint.h>

// ---- problem constants (match reference) ----
#define BB    4
#define NN    1000
#define CC    81
#define FG    (CC - 1)          // 80 foreground classes
#define DETS  100
#define SCORE_THRESH 0.05f
#define NMS_THRESH   0.5f
#define NEGV  (-1e30f)
#define PADK  (-3.0e38f)
#define BBOX_CLIP 4.135166556742356f   // log(1000/16)

typedef __attribute__((ext_vector_type(2))) float v2f;
typedef __attribute__((ext_vector_type(8))) float v8f;

// =====================================================================
// Kernel 1: softmax over C=81 classes. One wave (32 lanes) handles 16
// rows; row sums computed with V_WMMA_F32_16X16X4_F32 (f32 matrix pipe):
// D = A(16x4 exp tile) * ones(4x16) accumulated over 21 K-chunks -> every
// element of D row m equals rowsum(m), exact f32 math.
// =====================================================================
__global__ void __launch_bounds__(32)
softmax_kernel(const float* __restrict__ logits, float* __restrict__ probs) {
  __shared__ float e[16][84];   // exp values, padded to 84 (zeros beyond 81)
  __shared__ float rs[16];      // row sums

  const int lane = threadIdx.x;          // 0..31
  const int r    = lane & 15;            // row within tile
  const int half = lane >> 4;            // 0 / 1
  const int row  = blockIdx.x * 16 + r;  // 4000 rows = 250 blocks * 16

  const float* in = logits + (size_t)row * CC;
  const int c0 = half ? 41 : 0;
  const int c1 = half ? 81 : 41;

  float m = -3.4e38f;
  for (int c = c0; c < c1; ++c) m = fmaxf(m, in[c]);
  m = fmaxf(m, __shfl_xor(m, 16, 32));   // combine the two half-rows

  for (int c = c0; c < c1; ++c) e[r][c] = expf(in[c] - m);
  if (half) { e[r][81] = 0.f; e[r][82] = 0.f; e[r][83] = 0.f; }
  __syncthreads();

  // A layout (16x4 f32): lanes 0-15 hold K=0,1 for M=lane; lanes 16-31 hold
  // K=2,3 for M=lane-16.  B = all ones -> layout-independent.
  v8f acc = {};
  v2f bones = {1.0f, 1.0f};
  for (int ch = 0; ch < 21; ++ch) {      // 21*4 = 84 >= 81 (padded zeros)
    const int kb = ch * 4 + half * 2;
    v2f a = { e[r][kb], e[r][kb + 1] };
    acc = __builtin_amdgcn_wmma_f32_16x16x4_f32(
        false, a, false, bones, (short)0, acc, false, false);
  }
  // D layout: lane 0-15 vgpr v -> D[v][lane]; lane 16-31 vgpr v -> D[v+8][..]
  if (lane == 0)  { for (int v = 0; v < 8; ++v) rs[v]     = acc[v]; }
  if (lane == 16) { for (int v = 0; v < 8; ++v) rs[v + 8] = acc[v]; }
  __syncthreads();

  const float sum = rs[r];
  float* op = probs + (size_t)row * CC;
  for (int c = c0; c < c1; ++c) op[c] = e[r][c] / sum;
}

// =====================================================================
// Kernel 2: box decode + clip. One thread per (b,n,c) box.
// =====================================================================
__global__ void __launch_bounds__(256)
decode_kernel(const float* __restrict__ reg, const float* __restrict__ props,
              float* __restrict__ dec, const int* __restrict__ imw,
              const int* __restrict__ imh) {
  const int t = blockIdx.x * blockDim.x + threadIdx.x;
  if (t >= BB * NN * CC) return;
  const int c  = t % CC;
  const int bn = t / CC;
  const int n  = bn % NN;
  const int b  = bn / NN;

  const float* p = props + ((size_t)b * NN + n) * 4;
  const float x1 = p[0], y1 = p[1], x2 = p[2], y2 = p[3];
  const float w  = x2 - x1 + 1.0f;
  const float h  = y2 - y1 + 1.0f;
  const float cx = x1 + 0.5f * w;
  const float cy = y1 + 0.5f * h;

  const float* rr = reg + (size_t)bn * CC * 4 + (size_t)c * 4;
  const float dx = rr[0] * 0.1f;
  const float dy = rr[1] * 0.1f;
  const float dw = fminf(rr[2] * 0.2f, BBOX_CLIP);
  const float dh = fminf(rr[3] * 0.2f, BBOX_CLIP);

  const float pcx = dx * w + cx;
  const float pcy = dy * h + cy;
  const float pw  = expf(dw) * w;
  const float ph  = expf(dh) * h;

  const float W1 = (float)(*imw) - 1.0f;
  const float H1 = (float)(*imh) - 1.0f;

  float* o = dec + (size_t)t * 4;
  o[0] = fminf(fmaxf(pcx - 0.5f * pw, 0.f), W1);
  o[1] = fminf(fmaxf(pcy - 0.5f * ph, 0.f), H1);
  o[2] = fminf(fmaxf(pcx + 0.5f * pw - 1.0f, 0.f), W1);
  o[3] = fminf(fmaxf(pcy + 0.5f * ph - 1.0f, 0.f), H1);
}

// =====================================================================
// Kernel 3: per-(image,class) NMS. 320 workgroups x 256 threads.
// Boxes/scores staged into LDS via CDNA5 async global->LDS DMA
// (ASYNCcnt path), then bitonic sort of (score,idx) and greedy
// suppression over the K valid entries.
// =====================================================================
#define NMSB 256
__global__ void __launch_bounds__(NMSB)
nms_kernel(const float* __restrict__ probs, const float* __restrict__ dec,
           float* __restrict__ clsflat) {
  __shared__ float4 bx[NN];      // 16 KB  (decoded boxes, original order)
  __shared__ float  key[1024];   // scores (NEG for invalid), sorted
  __shared__ int    sidx[1024];  // original index, sorted alongside
  __shared__ int    sup[1024];   // suppression flags (sorted positions)
  __shared__ int    kcnt;

  const int tid = threadIdx.x;
  const int b   = blockIdx.x / FG;
  const int j   = blockIdx.x % FG + 1;   // class 1..80

  // ---- async DMA stage: boxes (B128) + raw scores (B32) into LDS ----
  for (int n = tid; n < NN; n += NMSB) {
    const uint64_t gb = (uint64_t)(uintptr_t)(dec + (((size_t)b * NN + n) * CC + j) * 4);
    const uint32_t lb = (uint32_t)(uintptr_t)&bx[n];
    asm volatile("global_load_async_to_lds_b128 %0, %1, off"
                 :: "v"(lb), "v"(gb) : "memory");
    const uint64_t gs = (uint64_t)(uintptr_t)(probs + ((size_t)b * NN + n) * CC + j);
    const uint32_t ls = (uint32_t)(uintptr_t)&key[n];
    asm volatile("global_load_async_to_lds_b32 %0, %1, off"
                 :: "v"(ls), "v"(gs) : "memory");
  }
  if (tid == 0) kcnt = 0;
  asm volatile("s_wait_asynccnt 0x0" ::: "memory");
  __syncthreads();

  // ---- init keys / indices / suppression flags ----
  for (int p = tid; p < 1024; p += NMSB) {
    if (p < NN) {
      const float s = key[p];
      key[p]  = (s > SCORE_THRESH) ? s : NEGV;
      sidx[p] = p;
    } else {
      key[p]  = PADK;    // below NEG -> sinks to the end
      sidx[p] = -1;
    }
    sup[p] = 0;
  }
  __syncthreads();

  // ---- bitonic sort descending on key (carry sidx) ----
  for (int k = 2; k <= 1024; k <<= 1) {
    for (int jj = k >> 1; jj > 0; jj >>= 1) {
      for (int i = tid; i < 1024; i += NMSB) {
        const int ixj = i ^ jj;
        if (ixj > i) {
          const bool desc = ((i & k) == 0);
          const float a = key[i], c = key[ixj];
          const bool sw = desc ? (a < c) : (a > c);
          if (sw) {
            key[i] = c; key[ixj] = a;
            const int t2 = sidx[i]; sidx[i] = sidx[ixj]; sidx[ixj] = t2;
          }
        }
      }
      __syncthreads();
    }
  }

  // ---- K = number of valid (sorted to the front) ----
  for (int p = tid; p < NN; p += NMSB)
    if (key[p] > SCORE_THRESH) atomicAdd(&kcnt, 1);
  __syncthreads();
  const int K = kcnt;

  // ---- greedy suppression (reference-faithful) ----
  for (int i = 0; i < K; ++i) {
    __syncthreads();
    if (sup[i]) continue;                 // uniform: all lanes read same LDS
    const float4 bi = bx[sidx[i]];
    const float ai = (bi.z - bi.x + 1.f) * (bi.w - bi.y + 1.f);
    for (int p = i + 1 + tid; p < K; p += NMSB) {
      if (sup[p]) continue;
      const float4 bj = bx[sidx[p]];
      const float aj = (bj.z - bj.x + 1.f) * (bj.w - bj.y + 1.f);
      const float xx1 = fmaxf(bi.x, bj.x);
      const float yy1 = fmaxf(bi.y, bj.y);
      const float xx2 = fminf(bi.z, bj.z);
      const float yy2 = fminf(bi.w, bj.w);
      const float iw = fmaxf(xx2 - xx1 + 1.f, 0.f);
      const float ih = fmaxf(yy2 - yy1 + 1.f, 0.f);
      const float inter = iw * ih;
      const float iou = inter / (ai + aj - inter);
      if (iou > NMS_THRESH) sup[p] = 1;
    }
  }
  __syncthreads();

  // ---- writeback: kept -> score, else NEG (every n covered once) ----
  float* out = clsflat + ((size_t)b * FG + (j - 1)) * NN;
  for (int p = tid; p < 1024; p += NMSB) {
    const int n = sidx[p];
    if (n >= 0) {
      const bool kept = (key[p] > SCORE_THRESH) && (sup[p] == 0);
      out[n] = kept ? key[p] : NEGV;
    }
  }
}

// =====================================================================
// Kernel 4: per-image top-100 via 100 block-wide argmax passes with
// smallest-index tie-break (matches jax.lax.top_k stable ordering).
// Destructively masks picks in clsflat (fully rewritten each launch).
// =====================================================================
__global__ void __launch_bounds__(256)
topk_kernel(float* __restrict__ clsflat, const float* __restrict__ dec,
            float* __restrict__ out) {
  __shared__ float bv[256];
  __shared__ int   bi[256];
  const int tid = threadIdx.x;
  const int b   = blockIdx.x;
  float* S = clsflat + (size_t)b * (FG * NN);

  for (int it = 0; it < DETS; ++it) {
    float best = -3.4e38f;
    int   bidx = 0x7fffffff;
    for (int i = tid; i < FG * NN; i += 256) {
      const float v = S[i];
      if (v > best || (v == best && i < bidx)) { best = v; bidx = i; }
    }
    bv[tid] = best; bi[tid] = bidx;
    __syncthreads();
    for (int s = 128; s > 0; s >>= 1) {
      if (tid < s) {
        const float ov = bv[tid + s];
        const int   oi = bi[tid + s];
        if (ov > bv[tid] || (ov == bv[tid] && oi < bi[tid])) {
          bv[tid] = ov; bi[tid] = oi;
        }
      }
      __syncthreads();
    }
    if (tid == 0) {
      const float sbest = bv[0];
      const int   ib    = bi[0];
      S[ib] = -3.4e38f;                      // remove from future passes
      const bool ok = sbest > (NEGV * 0.5f); // > -5e29
      const int cls = ib / NN;               // 0..79 -> label cls+1
      const int n   = ib % NN;
      const float* bp = dec + (((size_t)b * NN + n) * CC + (cls + 1)) * 4;
      float* dp = out + ((size_t)b * DETS + it) * 5;
      dp[0] = ok ? bp[0] : 0.f;
      dp[1] = ok ? bp[1] : 0.f;
      dp[2] = ok ? bp[2] : 0.f;
      dp[3] = ok ? bp[3] : 0.f;
      dp[4] = ok ? sbest : 0.f;
      out[(size_t)BB * DETS * 5 + (size_t)b * DETS + it] =
          ok ? (float)(cls + 1) : 0.f;
    }
    __syncthreads();
  }
}

// =====================================================================
extern "C" void kernel_launch(void* const* d_in, const int* in_sizes, int n_in,
                              void* d_out, int out_size, void* d_ws, size_t ws_size,
                              hipStream_t stream) {
  const float* logits = (const float*)d_in[0];
  const float* reg    = (const float*)d_in[1];
  const float* props  = (const float*)d_in[2];
  const int*   imw    = (const int*)d_in[3];
  const int*   imh    = (const int*)d_in[4];

  float* ws      = (float*)d_ws;
  float* probs   = ws;                               // 324000 f32
  float* dec     = ws + (size_t)BB * NN * CC;        // 1296000 f32
  float* clsflat = dec + (size_t)BB * NN * CC * 4;   // 320000 f32
  float* out     = (float*)d_out;

  softmax_kernel<<<(BB * NN) / 16, 32, 0, stream>>>(logits, probs);
  const int nbox = BB * NN * CC;
  decode_kernel<<<(nbox + 255) / 256, 256, 0, stream>>>(reg, props, dec, imw, imh);
  nms_kernel<<<BB * FG, NMSB, 0, stream>>>(probs, dec, clsflat);
  topk_kernel<<<BB, 256, 0, stream>>>(clsflat, dec, out);
}